// model1_51642686767567
// MI455X (gfx1250) — compile-verified
//
#include <hip/hip_runtime.h>
#include <hip/hip_bf16.h>

#define VOCAB 100000
#define EMB   128
#define BROWS 4096
#define SEQ   200

typedef __attribute__((ext_vector_type(2))) float v2f;
typedef __attribute__((ext_vector_type(8))) float v8f;

// ---------------------------------------------------------------------------
// Kernel 1: masked-mean pooling.  One block (8 waves / 256 threads) per row.
// Each wave strides over tokens; each lane owns a float4 slice of the 128-dim
// embedding, giving one coalesced 512B load per wave per valid token.
// Embedding table (51.2 MB) is L2-resident (192 MB L2), so the gather is
// L2-bandwidth bound after the first pass.
// ---------------------------------------------------------------------------
__global__ void pool_kernel(const int* __restrict__ x,
                            const float* __restrict__ emb,
                            float* __restrict__ pooled) {
    const int row  = blockIdx.x;        // 0..4095
    const int tid  = threadIdx.x;       // 0..255
    const int wv   = tid >> 5;          // wave in block: 0..7
    const int lane = tid & 31;

    __shared__ int    s_idx[SEQ];
    __shared__ float4 s_part[8][32];
    __shared__ float  s_cnt[8];

    // Stage token indices in LDS (int32; -1 marks padding)
    for (int s = tid; s < SEQ; s += 256)
        s_idx[s] = x[row * SEQ + s];
    __syncthreads();

    float4 acc = make_float4(0.f, 0.f, 0.f, 0.f);
    int cnt = 0;
    const float4* __restrict__ embv = (const float4*)emb;  // row stride = 32 float4

    for (int s = wv; s < SEQ; s += 8) {
        int idx = s_idx[s];            // wave-uniform
        if (idx >= 0) {
            float4 v = embv[(size_t)idx * 32 + lane];
            acc.x += v.x; acc.y += v.y; acc.z += v.z; acc.w += v.w;
            cnt++;
        }
    }

    s_part[wv][lane] = acc;
    if (lane == 0) s_cnt[wv] = (float)cnt;
    __syncthreads();

    if (wv == 0) {
        float4 t = s_part[0][lane];
        float  c = s_cnt[0];
        #pragma unroll
        for (int g = 1; g < 8; g++) {
            float4 p = s_part[g][lane];
            t.x += p.x; t.y += p.y; t.z += p.z; t.w += p.w;
            c += s_cnt[g];
        }
        float inv = 1.0f / c;          // col 0 always valid -> c >= 1
        float4 r = make_float4(t.x * inv, t.y * inv, t.z * inv, t.w * inv);
        ((float4*)pooled)[(size_t)row * 32 + lane] = r;
    }
}

// ---------------------------------------------------------------------------
// Kernel 2: out[4096,2] = pooled[4096,128] @ W^T[128,2] + b  via
// V_WMMA_F32_16X16X4_F32.  One wave per 16-row tile; N padded to 16
// (columns 2..15 of B are zeroed with a multiplicative mask, so the K-loop
// has NO divergent branches / EXEC churn -- just loads + wmma).
//
// 32-bit A 16x4 layout: lanes 0-15 -> K=0 (v0), K=1 (v1);
//                       lanes 16-31 -> K=2 (v0), K=3 (v1).
// B 4x16 mirrors it (row striped across lanes within a VGPR).
// C/D: lane l, vgpr v -> m = v + 8*(l>=16), n = l%16.
// Launch produces exactly BROWS/16 waves, so EXEC is all-ones throughout.
// ---------------------------------------------------------------------------
__global__ void proj_kernel(const float* __restrict__ pooled,
                            const float* __restrict__ W,
                            const float* __restrict__ bias,
                            float* __restrict__ out) {
    const int wave = (int)((blockIdx.x * blockDim.x + threadIdx.x) >> 5); // tile id
    const int lane = threadIdx.x & 31;

    const int m0   = wave * 16;
    const int half = lane >> 4;     // 0 or 1
    const int r    = lane & 15;     // row-in-tile for A, column n for B/C/D
    const int koff = half * 2;      // K sub-offset this half-wave owns

    // Branch-free B-fragment setup: clamp row (always in-bounds) + zero mask.
    const int   wrow = (r < 2) ? r : 0;
    const float msk  = (r < 2) ? 1.0f : 0.0f;
    const float* __restrict__ wptr = W + wrow * EMB + koff;         // + kb
    const float* __restrict__ aptr = pooled + (size_t)(m0 + r) * EMB + koff;

    v8f c = {};
    #pragma unroll 4
    for (int kb = 0; kb < EMB; kb += 4) {
        v2f a, bf;
        a.x  = aptr[kb];
        a.y  = aptr[kb + 1];
        bf.x = wptr[kb] * msk;      // B[k][n] = (n<2) ? W[n][k] : 0
        bf.y = wptr[kb + 1] * msk;
        c = __builtin_amdgcn_wmma_f32_16x16x4_f32(
                /*neg_a=*/false, a, /*neg_b=*/false, bf,
                /*c_mod=*/(short)0, c, /*reuse_a=*/false, /*reuse_b=*/false);
    }

    if (r < 2) {                    // only n = 0,1 are real outputs
        float bb = bias[r];
        #pragma unroll
        for (int v = 0; v < 8; v++) {
            int m = v + half * 8;
            out[(size_t)(m0 + m) * 2 + r] = c[v] + bb;
        }
    }
}

// ---------------------------------------------------------------------------
extern "C" void kernel_launch(void* const* d_in, const int* in_sizes, int n_in,
                              void* d_out, int out_size, void* d_ws, size_t ws_size,
                              hipStream_t stream) {
    (void)in_sizes; (void)n_in; (void)out_size; (void)ws_size;
    const int*   x    = (const int*)d_in[0];     // [4096,200] token ids (-1 = pad)
    const float* emb  = (const float*)d_in[1];   // [100000,128]
    const float* W    = (const float*)d_in[2];   // [2,128]
    const float* bias = (const float*)d_in[3];   // [2]
    float*       out  = (float*)d_out;           // [4096,2]
    float*       pooled = (float*)d_ws;          // [4096,128] scratch (2 MB)

    pool_kernel<<<BROWS, 256, 0, stream>>>(x, emb, pooled);

    // 256 tiles of 16 rows -> 256 waves -> 32 blocks x 8 waves (exact cover)
    proj_kernel<<<32, 256, 0, stream>>>(pooled, W, bias, out);
}